// IGMC_36240934044308
// MI455X (gfx1250) — compile-verified
//
#include <hip/hip_runtime.h>
#include <math.h>

#define NN      100000     // nodes (divisible by 16)
#define NE      3200000    // edges
#define NREL    5
#define NBASES  2
#define DD      32
#define WTOT    192        // 5*32 relation cols + 32 self-loop cols
#define NT      4096       // targets

typedef __attribute__((ext_vector_type(2))) float v2f;
typedef __attribute__((ext_vector_type(8))) float v8f;

// ---------------------------------------------------------------------------
// Build combined weight matrix Wbuf[K][192]:
//   cols 0..159  : W[r] = sum_b wcomp[r,b]*bases[b]   (r = col/32)
//   cols 160..191: w_self
// ---------------------------------------------------------------------------
__global__ void build_w_kernel(const float* __restrict__ bases,
                               const float* __restrict__ wcomp,
                               const float* __restrict__ wself,
                               float* __restrict__ Wbuf, int K) {
    int i = blockIdx.x * blockDim.x + threadIdx.x;
    if (i >= K * WTOT) return;
    int k = i / WTOT, c = i % WTOT;
    float v;
    if (c < NREL * DD) {
        int r = c >> 5, o = c & 31;
        v = 0.f;
        #pragma unroll
        for (int b = 0; b < NBASES; ++b)
            v += wcomp[r * NBASES + b] * bases[(b * K + k) * DD + o];
    } else {
        v = wself[k * DD + (c - NREL * DD)];
    }
    Wbuf[i] = v;
}

// ---------------------------------------------------------------------------
// hW[n, 0..191] = h[n, 0..K-1] @ Wbuf   via V_WMMA_F32_16X16X4_F32
// One wave (32 threads) per 16x16 output tile. All dims divide evenly,
// so EXEC is all-ones (WMMA requirement) with no bounds checks.
// ---------------------------------------------------------------------------
__global__ void gemm_hw_kernel(const float* __restrict__ h,
                               const float* __restrict__ Wbuf,
                               float* __restrict__ hW, int K) {
    int lane = threadIdx.x;          // 0..31, wave32
    int half = lane >> 4;            // 0 | 1
    int l15  = lane & 15;
    int row0 = blockIdx.x * 16;
    int col0 = blockIdx.y * 16;

    v8f acc = {};
    for (int kb = 0; kb < K; kb += 4) {
        int ka = kb + 2 * half;
        v2f a, b;
        // A 16x4: lanes0-15 K={kb,kb+1}; lanes16-31 K={kb+2,kb+3}; M=l15
        a.x = h[(row0 + l15) * K + ka];
        a.y = h[(row0 + l15) * K + ka + 1];
        // B 4x16: same K split; N=l15
        b.x = Wbuf[ka * WTOT + col0 + l15];
        b.y = Wbuf[(ka + 1) * WTOT + col0 + l15];
        acc = __builtin_amdgcn_wmma_f32_16x16x4_f32(
                false, a, false, b, (short)0, acc, false, false);
    }
    // D 16x16 f32: VGPR j -> M=j+8*half, N=l15
    #pragma unroll
    for (int j = 0; j < 8; ++j)
        hW[(row0 + j + 8 * half) * WTOT + col0 + l15] = acc[j];
}

__global__ void zero4_kernel(float4* __restrict__ p, int n4) {
    int i = blockIdx.x * blockDim.x + threadIdx.x;
    if (i < n4) p[i] = make_float4(0.f, 0.f, 0.f, 0.f);
}

// ---------------------------------------------------------------------------
// Edge scatter: agg[dst[e], f] += hW[src[e], etype[e]*32 + f]
// One thread per (edge, 4-feature chunk): a single b128 load from hW and
// 4 hardware fp32 atomics. agg (12.8MB) and hW (77MB) both fit in the
// 192MB L2, so the random traffic and atomics resolve in L2.
// ---------------------------------------------------------------------------
__global__ void edge_scatter_kernel(const int* __restrict__ src,
                                    const int* __restrict__ dst,
                                    const int* __restrict__ et,
                                    const float4* __restrict__ hW4,
                                    float* __restrict__ agg) {
    long long i = (long long)blockIdx.x * blockDim.x + threadIdx.x;
    if (i >= (long long)NE * (DD / 4)) return;
    int e = (int)(i >> 3);          // edge
    int c = (int)(i & 7);           // chunk of 4 features
    // hW row = src*192 floats = src*48 float4; rel block = et*8 float4
    float4 v = hW4[src[e] * (WTOT / 4) + et[e] * (DD / 4) + c];
    float* a = &agg[dst[e] * DD + c * 4];
    unsafeAtomicAdd(a + 0, v.x);
    unsafeAtomicAdd(a + 1, v.y);
    unsafeAtomicAdd(a + 2, v.z);
    unsafeAtomicAdd(a + 3, v.w);
}

// h_out[n, 4f..4f+3] = tanh(agg + hW_self + bias), vectorized by float4
__global__ void combine_kernel(const float4* __restrict__ agg4,
                               const float4* __restrict__ hW4,
                               const float* __restrict__ bias,
                               float4* __restrict__ hout4) {
    int i = blockIdx.x * blockDim.x + threadIdx.x;   // over NN * 8 chunks
    if (i >= NN * (DD / 4)) return;
    int n = i >> 3, c = i & 7;
    float4 a = agg4[i];
    float4 s = hW4[n * (WTOT / 4) + (NREL * DD) / 4 + c];
    const float* b = &bias[c * 4];
    float4 o;
    o.x = tanhf(a.x + s.x + b[0]);
    o.y = tanhf(a.y + s.y + b[1]);
    o.z = tanhf(a.z + s.z + b[2]);
    o.w = tanhf(a.w + s.w + b[3]);
    hout4[i] = o;
}

// z[t, 0:128] = cs[user_idx[t]], z[t,128:256] = cs[item_idx[t]]
// where cs[n] = [state0[n] | state1[n] | state2[n] | state3[n]]
__global__ void gather_z_kernel(const float* __restrict__ hstates,
                                const int* __restrict__ uidx,
                                const int* __restrict__ iidx,
                                float* __restrict__ z) {
    int i = blockIdx.x * blockDim.x + threadIdx.x;
    if (i >= NT * 256) return;
    int t = i >> 8, c = i & 255;
    int n  = (c < 128) ? uidx[t] : iidx[t];
    int cc = c & 127;
    int layer = cc >> 5, f = cc & 31;
    z[i] = hstates[((long long)layer * NN + n) * DD + f];
}

// z2 = relu(z @ lin1_w + lin1_b): M=4096, K=256, N=128 via WMMA f32
__global__ void lin1_gemm_kernel(const float* __restrict__ z,
                                 const float* __restrict__ w,
                                 const float* __restrict__ b,
                                 float* __restrict__ z2) {
    int lane = threadIdx.x;
    int half = lane >> 4;
    int l15  = lane & 15;
    int row0 = blockIdx.x * 16;
    int col0 = blockIdx.y * 16;

    v8f acc = {};
    for (int kb = 0; kb < 256; kb += 4) {
        int ka = kb + 2 * half;
        v2f a, bb;
        a.x  = z[(row0 + l15) * 256 + ka];
        a.y  = z[(row0 + l15) * 256 + ka + 1];
        bb.x = w[ka * 128 + col0 + l15];
        bb.y = w[(ka + 1) * 128 + col0 + l15];
        acc = __builtin_amdgcn_wmma_f32_16x16x4_f32(
                false, a, false, bb, (short)0, acc, false, false);
    }
    float bv = b[col0 + l15];
    #pragma unroll
    for (int j = 0; j < 8; ++j) {
        float v = acc[j] + bv;
        z2[(row0 + j + 8 * half) * 128 + col0 + l15] = v > 0.f ? v : 0.f;
    }
}

// out[t] = z2[t,:] . lin2_w + lin2_b
__global__ void lin2_dot_kernel(const float* __restrict__ z2,
                                const float* __restrict__ w,
                                const float* __restrict__ b,
                                float* __restrict__ out) {
    int t = blockIdx.x * blockDim.x + threadIdx.x;
    if (t >= NT) return;
    float s = b[0];
    #pragma unroll 4
    for (int f = 0; f < 128; ++f) s += z2[t * 128 + f] * w[f];
    out[t] = s;
}

extern "C" void kernel_launch(void* const* d_in, const int* in_sizes, int n_in,
                              void* d_out, int out_size, void* d_ws, size_t ws_size,
                              hipStream_t stream) {
    const float* x      = (const float*)d_in[0];
    const int*   src    = (const int*)d_in[1];
    const int*   dst    = (const int*)d_in[2];
    const int*   et     = (const int*)d_in[3];
    const int*   uidx   = (const int*)d_in[4];
    const int*   iidx   = (const int*)d_in[5];
    const float* bases0 = (const float*)d_in[6];
    const float* wcomp0 = (const float*)d_in[7];
    const float* wself0 = (const float*)d_in[8];
    const float* bias0  = (const float*)d_in[9];
    const float* basesR = (const float*)d_in[10];
    const float* wcompR = (const float*)d_in[11];
    const float* wselfR = (const float*)d_in[12];
    const float* biasR  = (const float*)d_in[13];
    const float* lin1w  = (const float*)d_in[14];
    const float* lin1b  = (const float*)d_in[15];
    const float* lin2w  = (const float*)d_in[16];
    const float* lin2b  = (const float*)d_in[17];
    float* out = (float*)d_out;
    float* ws  = (float*)d_ws;

    // workspace layout (floats)
    float* hstates = ws;                       // 4 * NN * 32  = 12,800,000
    float* hW      = ws + 12800000;            // NN * 192     = 19,200,000
    float* agg     = ws + 32000000;            // NN * 32      =  3,200,000
    float* Wbuf    = ws + 35200000;            // 32 * 192     =      6,144
    float* z       = hW;                       // reuse after layers: 4096*256
    float* z2      = hW + NT * 256;            // 4096*128

    for (int l = 0; l < 4; ++l) {
        const int    K     = (l == 0) ? 4 : DD;
        const float* bases = (l == 0) ? bases0 : basesR + (size_t)(l - 1) * NBASES * DD * DD;
        const float* wc    = (l == 0) ? wcomp0 : wcompR + (size_t)(l - 1) * NREL * NBASES;
        const float* wsf   = (l == 0) ? wself0 : wselfR + (size_t)(l - 1) * DD * DD;
        const float* bia   = (l == 0) ? bias0  : biasR  + (size_t)(l - 1) * DD;
        const float* hin   = (l == 0) ? x      : hstates + (size_t)(l - 1) * NN * DD;
        float*       hout  = hstates + (size_t)l * NN * DD;

        build_w_kernel<<<(K * WTOT + 255) / 256, 256, 0, stream>>>(bases, wc, wsf, Wbuf, K);
        gemm_hw_kernel<<<dim3(NN / 16, WTOT / 16), 32, 0, stream>>>(hin, Wbuf, hW, K);
        zero4_kernel<<<(NN * DD / 4 + 255) / 256, 256, 0, stream>>>((float4*)agg, NN * DD / 4);
        long long tot = (long long)NE * (DD / 4);
        edge_scatter_kernel<<<(unsigned)((tot + 255) / 256), 256, 0, stream>>>(
            src, dst, et, (const float4*)hW, agg);
        combine_kernel<<<(NN * DD / 4 + 255) / 256, 256, 0, stream>>>(
            (const float4*)agg, (const float4*)hW, bia, (float4*)hout);
    }

    gather_z_kernel<<<(NT * 256 + 255) / 256, 256, 0, stream>>>(hstates, uidx, iidx, z);
    lin1_gemm_kernel<<<dim3(NT / 16, 128 / 16), 32, 0, stream>>>(z, lin1w, lin1b, z2);
    lin2_dot_kernel<<<(NT + 255) / 256, 256, 0, stream>>>(z2, lin2w, lin2b, out);
}